// ModularDrumSynth_27565100106266
// MI455X (gfx1250) — compile-verified
//
#include <hip/hip_runtime.h>
#include <math.h>
#include <stdint.h>

#define MAX_DELAY 1000
#define BLOCK 256
#define EPT 4
#define TILE (BLOCK * EPT)
#define LOG2E 1.44269504088896340736f

typedef float v4f __attribute__((ext_vector_type(4)));

// CDNA5 async global->LDS path (ASYNCcnt-tracked DMA into LDS).
#if defined(__gfx1250__) && __has_builtin(__builtin_amdgcn_global_load_async_to_lds_b32)
#define USE_ASYNC_LDS 1
#else
#define USE_ASYNC_LDS 0
#endif

__device__ __forceinline__ float fast_exp2(float x) {
#if __has_builtin(__builtin_amdgcn_exp2f)
    return __builtin_amdgcn_exp2f(x);   // v_exp_f32
#else
    return exp2f(x);
#endif
}

// d_ws float layout:
// [0]=res_decay [1]=log2(res_decay) [2]=trans_gain [3]=noise_gain
// [4]=a1 (per-sample log2-step, attack)  [5]=s1=exp2(a1)
// [6]=a2 (decay)                          [7]=s2=exp2(a2)
// [8]=a3 (noise decay)                    [9]=s3=exp2(a3)
__global__ void drum_setup(const float* __restrict__ rd_p,
                           const float* __restrict__ ta_p,
                           const float* __restrict__ td_p,
                           const float* __restrict__ tg_p,
                           const float* __restrict__ ng_p,
                           const float* __restrict__ nd_p,
                           float inv, float* __restrict__ ws) {
    float rd = *rd_p;
    float a1 = -LOG2E * inv / *ta_p;
    float a2 = -LOG2E * inv / *td_p;
    float a3 = -LOG2E * inv / *nd_p;
    ws[0] = rd;
    ws[1] = log2f(rd);
    ws[2] = *tg_p;
    ws[3] = *ng_p;
    ws[4] = a1; ws[5] = fast_exp2(a1);
    ws[6] = a2; ws[7] = fast_exp2(a2);
    ws[8] = a3; ws[9] = fast_exp2(a3);
}

__global__ void __launch_bounds__(BLOCK) drum_main(
    const float* __restrict__ res_buffer,
    const float* __restrict__ noise1,
    const float* __restrict__ noise2,
    const float* __restrict__ ws,
    float* __restrict__ out,
    int N)
{
    __shared__ float s_res[MAX_DELAY];

    // Stage the 4KB comb-filter delay line into LDS once per workgroup.
#if USE_ASYNC_LDS
    for (int i = threadIdx.x; i < MAX_DELAY; i += BLOCK) {
        __builtin_amdgcn_global_load_async_to_lds_b32(
            (__attribute__((address_space(1))) int*)(uintptr_t)(res_buffer + i),
            (__attribute__((address_space(3))) int*)(&s_res[i]),
            /*offset=*/0, /*cpol=*/0);
    }
#if __has_builtin(__builtin_amdgcn_s_wait_asynccnt)
    __builtin_amdgcn_s_wait_asynccnt(0);
#else
    asm volatile("s_wait_asynccnt 0" ::: "memory");
#endif
#else
    for (int i = threadIdx.x; i < MAX_DELAY; i += BLOCK) s_res[i] = res_buffer[i];
#endif
    __syncthreads();

    const float rd = ws[0], l2rd = ws[1];
    const float tg = ws[2], ng   = ws[3];
    const float a1 = ws[4], s1 = ws[5];
    const float a2 = ws[6], s2 = ws[7];
    const float a3 = ws[8], s3 = ws[9];

    int n0 = (int)(blockIdx.x * BLOCK + threadIdx.x) * EPT;
    if (n0 >= N) return;

    // Comb term: res_decay^(n/1000 + 1), constant within the thread except at a
    // multiple-of-1000 boundary (handled with one multiply by rd).
    unsigned q = (unsigned)n0 / MAX_DELAY;
    int r = n0 - (int)q * MAX_DELAY;
    float p = fast_exp2((float)(q + 1) * l2rd);

    // Envelopes: exp2(n * a) with geometric per-sample advance s = exp2(a).
    float nf = (float)n0;
    float e1 = fast_exp2(nf * a1);
    float e2 = fast_exp2(nf * a2);
    float e3 = fast_exp2(nf * a3);

    if (n0 + EPT <= N) {
        v4f z1 = __builtin_nontemporal_load((const v4f*)(noise1 + n0));
        v4f z2 = __builtin_nontemporal_load((const v4f*)(noise2 + n0));
        v4f o;
#pragma unroll
        for (int j = 0; j < EPT; ++j) {
            float res = p * s_res[r];
            o[j] = fmaf(tg * (e1 - e2), z1[j], fmaf(ng * e3, z2[j], res));
            ++r;
            if (r == MAX_DELAY) { r = 0; p *= rd; }
            e1 *= s1; e2 *= s2; e3 *= s3;
        }
        __builtin_nontemporal_store(o, (v4f*)(out + n0));
    } else {
        for (int j = 0; j < EPT && (n0 + j) < N; ++j) {
            float res = p * s_res[r];
            out[n0 + j] = fmaf(tg * (e1 - e2), noise1[n0 + j],
                               fmaf(ng * e3, noise2[n0 + j], res));
            ++r;
            if (r == MAX_DELAY) { r = 0; p *= rd; }
            e1 *= s1; e2 *= s2; e3 *= s3;
        }
    }
}

extern "C" void kernel_launch(void* const* d_in, const int* in_sizes, int n_in,
                              void* d_out, int out_size, void* d_ws, size_t ws_size,
                              hipStream_t stream) {
    const float* res_decay    = (const float*)d_in[0];
    const float* res_buffer   = (const float*)d_in[1];
    const float* trans_attack = (const float*)d_in[2];
    const float* trans_decay  = (const float*)d_in[3];
    const float* trans_gain   = (const float*)d_in[4];
    const float* noise_gain   = (const float*)d_in[5];
    const float* noise_decay  = (const float*)d_in[6];
    const float* noise1       = (const float*)d_in[7];
    const float* noise2       = (const float*)d_in[8];
    (void)in_sizes; (void)n_in; (void)ws_size;

    float* out = (float*)d_out;
    float* ws  = (float*)d_ws;
    int N = out_size;
    float inv = (N > 1) ? 1.0f / (float)(N - 1) : 0.0f;

    drum_setup<<<1, 1, 0, stream>>>(res_decay, trans_attack, trans_decay,
                                    trans_gain, noise_gain, noise_decay, inv, ws);

    int grid = (N + TILE - 1) / TILE;
    drum_main<<<grid, BLOCK, 0, stream>>>(res_buffer, noise1, noise2, ws, out, N);
}